// TPMoEWrapper_63324997812518
// MI455X (gfx1250) — compile-verified
//
#include <hip/hip_runtime.h>
#include <math.h>

#define NE     8
#define TOPK   2
#define D_DIM  512
#define F_DIM  2048
#define T_TOK  4096
#define FC     128     // F chunk per stage-1/2 round
#define NW     8       // waves per block (256 threads, wave32)

typedef __attribute__((ext_vector_type(2))) float v2f;
typedef __attribute__((ext_vector_type(8))) float v8f;

// D = A(16x4,f32) * B(4x16,f32) + C(16x16,f32)
__device__ __forceinline__ v8f wmma_f32(v2f a, v2f b, v8f c) {
    return __builtin_amdgcn_wmma_f32_16x16x4_f32(false, a, false, b, (short)0, c,
                                                 false, false);
}

// ---------------------------------------------------------------------------
// Kernel 0: zero the dense output region and the expert counters.
// ---------------------------------------------------------------------------
__global__ void moe_zero_kernel(float* __restrict__ out, int n, int* __restrict__ cnt) {
    int i = blockIdx.x * blockDim.x + threadIdx.x;
    for (int j = i; j < n; j += gridDim.x * blockDim.x) out[j] = 0.0f;
    if (i < NE) cnt[i] = 0;
}

// ---------------------------------------------------------------------------
// Kernel 1: router. One wave32 per token: logits = x[t] . gate_w[e], top-2,
// softmax over the two, scatter (token, weight) into per-expert lists.
// ---------------------------------------------------------------------------
__global__ void __launch_bounds__(256)
moe_router_kernel(const float* __restrict__ x, const float* __restrict__ gw,
                  float* __restrict__ logits_out, int* __restrict__ cnt,
                  int* __restrict__ tok, float* __restrict__ wts) {
    const int lane = threadIdx.x & 31;
    const int wave = threadIdx.x >> 5;
    const int t    = blockIdx.x * NW + wave;
    if (t >= T_TOK) return;

    float acc[NE];
#pragma unroll
    for (int e = 0; e < NE; ++e) acc[e] = 0.0f;

    const float* xr = x + (size_t)t * D_DIM;
#pragma unroll 4
    for (int j = 0; j < D_DIM / 32; ++j) {
        float xv = xr[lane + 32 * j];
#pragma unroll
        for (int e = 0; e < NE; ++e) acc[e] += xv * gw[e * D_DIM + lane + 32 * j];
    }
#pragma unroll
    for (int e = 0; e < NE; ++e) {
        float v = acc[e];
        for (int off = 16; off; off >>= 1) v += __shfl_xor(v, off, 32);
        acc[e] = v;
    }

    if (lane == 0) {
        float v0 = -1.0e30f, v1 = -1.0e30f;
        int   i0 = 0, i1 = 0;
#pragma unroll
        for (int e = 0; e < NE; ++e) {
            float l = acc[e];
            logits_out[(size_t)t * NE + e] = l;
            if (l > v0)      { v1 = v0; i1 = i0; v0 = l; i0 = e; }
            else if (l > v1) { v1 = l;  i1 = e; }
        }
        float ex = expf(v1 - v0);          // softmax over the selected pair
        float w0 = 1.0f / (1.0f + ex);
        float w1 = ex   / (1.0f + ex);
        int s0 = atomicAdd(&cnt[i0], 1);
        tok[i0 * T_TOK + s0] = t;  wts[i0 * T_TOK + s0] = w0;
        int s1 = atomicAdd(&cnt[i1], 1);
        tok[i1 * T_TOK + s1] = t;  wts[i1 * T_TOK + s1] = w1;
    }
}

// ---------------------------------------------------------------------------
// Kernel 2: expert FFN on gathered token tiles using V_WMMA_F32_16X16X4_F32.
// Block = (expert e, 16-token tile). 8 waves:
//   stage 1: wave w makes h tile for f-cols [fc + 16w, +16):
//            h = silu(X W1^T) * (X W3^T), X from LDS, acc f32 in WMMA.
//   stage 2: wave w accumulates y[:, 64w : 64w+64] += h_chunk @ W2_chunk^T.
// Final: out[t] += weight[t] * y  via f32 global atomics.
// ---------------------------------------------------------------------------
__global__ void __launch_bounds__(256)
moe_ffn_kernel(const float* __restrict__ x,  const float* __restrict__ w1,
               const float* __restrict__ w2, const float* __restrict__ w3,
               const int* __restrict__ cnt,  const int* __restrict__ tok,
               const float* __restrict__ wts, float* __restrict__ out) {
    __shared__ float sX[16][516];   // padded stride: conflict-free A reads
    __shared__ float sH[16][132];
    __shared__ int   sTok[16];
    __shared__ float sWt[16];

    const int e  = blockIdx.x >> 8;
    const int mt = blockIdx.x & 255;
    const int ne = cnt[e];
    if (mt * 16 >= ne) return;      // uniform early exit, before any WMMA

    const int tid  = threadIdx.x;
    const int lane = tid & 31;
    const int wave = tid >> 5;
    const int hi   = lane >> 4;     // half-wave select (K/M split per ISA layout)
    const int nn   = lane & 15;

    if (tid < 16) {
        int idx = mt * 16 + tid;
        if (idx < ne) { sTok[tid] = tok[e * T_TOK + idx]; sWt[tid] = wts[e * T_TOK + idx]; }
        else          { sTok[tid] = -1;                   sWt[tid] = 0.0f; }
    }
    __syncthreads();

    // Gather X tile (16 x 512) into LDS, coalesced.
    for (int i = tid; i < 16 * D_DIM; i += 256) {
        int r = i >> 9, c = i & 511;
        int tk = sTok[r];
        sX[r][c] = (tk >= 0) ? x[(size_t)tk * D_DIM + c] : 0.0f;
    }
    __syncthreads();

    const float* W1e = w1 + (size_t)e * F_DIM * D_DIM;
    const float* W3e = w3 + (size_t)e * F_DIM * D_DIM;
    const float* W2e = w2 + (size_t)e * D_DIM * F_DIM;

    v8f yacc[4];
#pragma unroll
    for (int i = 0; i < 4; ++i) yacc[i] = (v8f){0, 0, 0, 0, 0, 0, 0, 0};

    for (int fc = 0; fc < F_DIM; fc += FC) {
        // ---- stage 1: h tile (16 tokens x 16 f-cols) for this wave ----
        const int    fb   = fc + wave * 16;
        const float* b1p  = W1e + (size_t)(fb + nn) * D_DIM + 2 * hi; // B row = W row nn
        const float* b3p  = W3e + (size_t)(fb + nn) * D_DIM + 2 * hi;
        v8f a1 = (v8f){0, 0, 0, 0, 0, 0, 0, 0};
        v8f a3 = (v8f){0, 0, 0, 0, 0, 0, 0, 0};
#pragma unroll 4
        for (int k = 0; k < D_DIM; k += 4) {
            v2f av = *(const v2f*)&sX[nn][k + 2 * hi];   // A: M=nn, K=k+2hi..+1
            v2f b1 = *(const v2f*)(b1p + k);
            v2f b3 = *(const v2f*)(b3p + k);
            a1 = wmma_f32(av, b1, a1);
            a3 = wmma_f32(av, b3, a3);
        }
        // silu(a1) * a3, store to LDS h chunk (C/D layout: row = j + 8*hi, col = nn)
#pragma unroll
        for (int j = 0; j < 8; ++j) {
            float z = a1[j];
            float s = z / (1.0f + expf(-z));
            sH[j + 8 * hi][wave * 16 + nn] = s * a3[j];
        }
        __syncthreads();

        // ---- stage 2: y[:, 64*wave .. +64) += h_chunk @ W2_chunk^T ----
#pragma unroll
        for (int dt = 0; dt < 4; ++dt) {
            const int    db  = wave * 64 + dt * 16;
            const float* b2p = W2e + (size_t)(db + nn) * F_DIM + fc + 2 * hi;
            v8f acc = yacc[dt];
#pragma unroll 4
            for (int k = 0; k < FC; k += 4) {
                v2f av = *(const v2f*)&sH[nn][k + 2 * hi];
                v2f bv = *(const v2f*)(b2p + k);
                acc = wmma_f32(av, bv, acc);
            }
            yacc[dt] = acc;
        }
        __syncthreads();            // sH reused next chunk
    }

    // Weighted combine into out (2 experts per token -> atomic f32 add).
#pragma unroll
    for (int dt = 0; dt < 4; ++dt) {
#pragma unroll
        for (int j = 0; j < 8; ++j) {
            int m  = j + 8 * hi;
            int tk = sTok[m];
            if (tk >= 0)
                atomicAdd(&out[(size_t)tk * D_DIM + wave * 64 + dt * 16 + nn],
                          sWt[m] * yacc[dt][j]);
        }
    }
}

// ---------------------------------------------------------------------------
extern "C" void kernel_launch(void* const* d_in, const int* in_sizes, int n_in,
                              void* d_out, int out_size, void* d_ws, size_t ws_size,
                              hipStream_t stream) {
    const float* x   = (const float*)d_in[0];   // [T, D]
    const float* gw  = (const float*)d_in[1];   // [E, D]
    const float* w1  = (const float*)d_in[2];   // [E, F, D]
    const float* w2  = (const float*)d_in[3];   // [E, D, F]
    const float* w3  = (const float*)d_in[4];   // [E, F, D]

    float* out    = (float*)d_out;              // [T, D]
    float* logits = out + (size_t)T_TOK * D_DIM; // [T, E]

    int*   cnt = (int*)d_ws;                    // 8 counters (16-slot padded)
    int*   tok = cnt + 16;                      // [E, T]
    float* wts = (float*)(tok + NE * T_TOK);    // [E, T]

    moe_zero_kernel<<<1024, 256, 0, stream>>>(out, T_TOK * D_DIM, cnt);
    moe_router_kernel<<<T_TOK / NW, 256, 0, stream>>>(x, gw, logits, cnt, tok, wts);
    moe_ffn_kernel<<<NE * 256, 256, 0, stream>>>(x, w1, w2, w3, cnt, tok, wts, out);
}